// CormorantAtomLevel_2954937500424
// MI455X (gfx1250) — compile-verified
//
#include <hip/hip_runtime.h>
#include <math.h>
#include <stdint.h>

typedef __attribute__((ext_vector_type(2))) float v2f;
typedef __attribute__((ext_vector_type(8))) float v8f;

// ---------------- shared host/device constants ----------------
constexpr int cW[4]   = {16, 39, 51, 50};    // packed agg output width per l1
constexpr int cWp[4]  = {16, 48, 64, 64};    // padded to 16
constexpr int cNTw[4] = {1, 3, 4, 4};        // n-tiles
constexpr int cNP[4]  = {4, 9, 11, 10};      // n_pairs(l)
constexpr int cT[4]   = {576, 1216, 1472, 1344}; // (2*np+1)*64
constexpr int cQoff[4] = {0, 1, 4, 9};       // offsets of l-blocks in 16-wide atom concat
constexpr int colL2tab[16] = {0,1,1,1,2,2,2,2,2,3,3,3,3,3,3,3};
constexpr size_t OUT_OFF[4] = {0, 49152, 196608, 442368};

// workspace layout (units: floats)
constexpr size_t AGG_COEF = 0;        // [4][64][8] f
constexpr size_t AGG_Q    = 2048;     // [4][64][8] i
constexpr size_t AGG_META = 4096;     // [4][64][4] i (slot,l,m,-)
constexpr size_t SQ_COEF  = 5120;     // [34][7][8] f (padded 2048)
constexpr size_t SQ_Q2    = 7168;     // [34][7][8] i (padded 2048)
constexpr size_t SQ_META  = 9216;     // [34][4] i (padded 1024)
constexpr size_t CAT_BASE = 10240;
constexpr size_t CATR[4] = {CAT_BASE, CAT_BASE+221184, CAT_BASE+1622016, CAT_BASE+4448256};
constexpr size_t CATI[4] = {CAT_BASE+8060928, CAT_BASE+8282112, CAT_BASE+9682944, CAT_BASE+12509184};
constexpr size_t WS_FLOATS = CAT_BASE + 2*8060928;

__host__ __device__ inline int imin_(int a,int b){return a<b?a:b;}
__host__ __device__ inline int iabs_(int a){return a<0?-a:a;}

// LDS byte offset from a generic pointer to __shared__ (aperture: low 32 bits = LDS offset)
__device__ inline unsigned ldsOff_(const void* p){ return (unsigned)(uintptr_t)p; }

// ---------------- init kernel: CG plan tables ----------------
__device__ double dfact_(int n){
  const double f[11] = {1.,1.,2.,6.,24.,120.,720.,5040.,40320.,362880.,3628800.};
  return f[n];
}
__device__ float cgcoef_(int l1,int m1,int l2,int m2,int l,int m){
  if (m != m1+m2) return 0.f;
  double pre = sqrt((2.0*l+1.0)*dfact_(l+l1-l2)*dfact_(l-l1+l2)*dfact_(l1+l2-l)/dfact_(l1+l2+l+1));
  pre *= sqrt(dfact_(l+m)*dfact_(l-m)*dfact_(l1-m1)*dfact_(l1+m1)*dfact_(l2-m2)*dfact_(l2+m2));
  int kmin = 0; if (l2-l-m1 > kmin) kmin = l2-l-m1; if (l1-l+m2 > kmin) kmin = l1-l+m2;
  int kmax = l1+l2-l; if (l1-m1 < kmax) kmax = l1-m1; if (l2+m2 < kmax) kmax = l2+m2;
  double s = 0.0;
  for (int k=kmin;k<=kmax;++k){
    double term = 1.0/(dfact_(k)*dfact_(l1+l2-l-k)*dfact_(l1-m1-k)*dfact_(l2+m2-k)*dfact_(l-l2+m1+k)*dfact_(l-l1-m2+k));
    s += (k & 1) ? -term : term;
  }
  return (float)(pre*s);
}
__device__ int slotIndex_(int l,int L1,int L2){
  int s = 0;
  for (int a=0;a<=3;++a) for (int b2=0;b2<=3;++b2){
    int lo = iabs_(a-b2), hi = imin_(a+b2,3);
    if (l>=lo && l<=hi){ if (a==L1 && b2==L2) return s; s++; }
  }
  return 0;
}

__global__ void init_plan_kernel(float* __restrict__ ws){
  if (threadIdx.x != 0 || blockIdx.x != 0) return;
  float* aggCoef = ws + AGG_COEF;
  int*   aggQ    = (int*)(ws + AGG_Q);
  int*   aggMeta = (int*)(ws + AGG_META);
  for (int l1=0;l1<4;l1++){
    int wdx = 0;
    for (int l2=0;l2<4;l2++){
      int lo = iabs_(l1-l2), hi = imin_(l1+l2,3);
      for (int l=lo;l<=hi;l++){
        int slot = slotIndex_(l,l1,l2);
        for (int mi=0;mi<2*l+1;mi++){
          int* mp = aggMeta + ((size_t)l1*64+wdx)*4;
          mp[0]=slot; mp[1]=l; mp[2]=mi; mp[3]=0;
          for (int p=0;p<8;p++){
            float cf = 0.f; int q = 0;
            if (p < 2*l1+1){
              int m1 = p-l1, m = mi-l, m2 = m-m1;
              if (m2 >= -l2 && m2 <= l2){ cf = cgcoef_(l1,m1,l2,m2,l,m); q = cQoff[l2]+m2+l2; }
            }
            aggCoef[((size_t)l1*64+wdx)*8+p] = cf;
            aggQ[((size_t)l1*64+wdx)*8+p]    = q;
          }
          wdx++;
        }
      }
    }
  }
  float* sqCoef = ws + SQ_COEF;
  int*   sqQ2   = (int*)(ws + SQ_Q2);
  int*   sqMeta = (int*)(ws + SQ_META);
  for (int k=0;k<2048;k++){ sqCoef[k]=0.f; sqQ2[k]=0; }
  int ci = 0;
  for (int l1=0;l1<4;l1++) for (int l2=0;l2<4;l2++){
    int lo = iabs_(l1-l2), hi = imin_(l1+l2,3);
    for (int l=lo;l<=hi;l++){
      sqMeta[ci*4+0]=l; sqMeta[ci*4+1]=slotIndex_(l,l1,l2); sqMeta[ci*4+2]=l1; sqMeta[ci*4+3]=l2;
      for (int mi=0;mi<2*l+1;mi++) for (int p=0;p<2*l1+1;p++){
        int m = mi-l, m1 = p-l1, m2 = m-m1;
        if (m2 >= -l2 && m2 <= l2){
          sqCoef[((size_t)ci*7+mi)*8+p] = cgcoef_(l1,m1,l2,m2,l,m);
          sqQ2[((size_t)ci*7+mi)*8+p]   = cQoff[l2]+m2+l2;
        }
      }
      ci++;
    }
  }
}

// ---------------- reps_sq + atom identity -> cat ----------------
__global__ __launch_bounds__(256) void sq_kernel(
    const float* __restrict__ A0, const float* __restrict__ A1,
    const float* __restrict__ A2, const float* __restrict__ A3,
    float* __restrict__ ws)
{
  int gid = blockIdx.x*256 + threadIdx.x;      // B*N*C = 24576 threads
  int b = gid / (96*64); int rem = gid - b*96*64; int i = rem >> 6; int c = rem & 63;
  const float* A[4] = {A0,A1,A2,A3};
  float ar[16], ai[16];
  for (int col=0; col<16; ++col){
    int l2 = colL2tab[col]; int q = col - cQoff[l2];
    const float* ap = A[l2] + ((((size_t)b*96 + i)*64 + c)*(2*l2+1) + q)*2;
    ar[col] = ap[0]; ai[col] = ap[1];
  }
  // identity block
  for (int l=0;l<4;l++){
    size_t t = (size_t)cNP[l]*64 + c;
    for (int mi=0; mi<2*l+1; ++mi){
      size_t base = (((size_t)b*cT[l] + t)*(2*l+1) + mi)*96 + i;
      ws[CATR[l]+base] = ar[cQoff[l]+mi];
      ws[CATI[l]+base] = ai[cQoff[l]+mi];
    }
  }
  // self CG products
  const int*   sqMeta = (const int*)(ws + SQ_META);
  const float* sqCoef = ws + SQ_COEF;
  const int*   sqQ2   = (const int*)(ws + SQ_Q2);
  for (int ci=0; ci<34; ++ci){
    int l = sqMeta[ci*4+0], slot = sqMeta[ci*4+1], L1 = sqMeta[ci*4+2];
    size_t t = (size_t)(cNP[l] + 1 + slot)*64 + c;
    for (int mi=0; mi<2*l+1; ++mi){
      float zr = 0.f, zi = 0.f;
      for (int p=0;p<2*L1+1;++p){
        float cf = sqCoef[((size_t)ci*7+mi)*8+p];
        if (cf != 0.f){
          int q1 = cQoff[L1]+p, q2 = sqQ2[((size_t)ci*7+mi)*8+p];
          float xr=ar[q1], xi=ai[q1], yr=ar[q2], yi=ai[q2];
          zr += cf*(xr*yr - xi*yi);
          zi += cf*(xr*yi + xi*yr);
        }
      }
      size_t base = (((size_t)b*cT[l] + t)*(2*l+1) + mi)*96 + i;
      ws[CATR[l]+base] = zr;
      ws[CATI[l]+base] = zi;
    }
  }
}

// ---------------- aggregation: complex GEMM with WMMA f32 16x16x4 ----------------
// Edge chunk staged via GLOBAL_LOAD_ASYNC_TO_LDS_B64 (ASYNCcnt path), overlapped
// with the T-chunk VALU build; completed with s_wait_asynccnt before the barrier.
template<int L1>
__global__ __launch_bounds__(256) void agg_kernel(
    const float* __restrict__ E,
    const float* __restrict__ A0, const float* __restrict__ A1,
    const float* __restrict__ A2, const float* __restrict__ A3,
    float* __restrict__ ws)
{
  constexpr int P  = 2*L1+1;
  constexpr int K  = 96*P;
  constexpr int W  = cW[L1];
  constexpr int Wp = cWp[L1];
  constexpr int NT = cNTw[L1];
  constexpr int TOT = 6*NT;

  __shared__ float  aR[96*17], aI[96*17];   // atoms, 16-wide l2-concat, pitch 17
  __shared__ float2 e2[96*17];              // edge chunk 96 x 16 interleaved (re,im), pitch 17
  __shared__ float  tR[16*65], tI[16*65];   // T chunk 16 x Wp, pitch 65

  const int tid = threadIdx.x;
  const int b = blockIdx.x >> 6;
  const int c = blockIdx.x & 63;
  const float* A[4] = {A0,A1,A2,A3};

  for (int idx = tid; idx < 96*16; idx += 256){
    int j = idx >> 4, col = idx & 15;
    int l2 = colL2tab[col]; int q = col - cQoff[l2];
    const float* ap = A[l2] + ((((size_t)b*96 + j)*64 + c)*(2*l2+1) + q)*2;
    aR[j*17+col] = ap[0]; aI[j*17+col] = ap[1];
  }

  const float* aggCoef = ws + AGG_COEF + (size_t)L1*64*8;
  const int*   aggQ    = (const int*)(ws + AGG_Q) + L1*64*8;
  const int*   aggMeta = (const int*)(ws + AGG_META) + L1*64*4;

  const int wave = tid >> 5, lane = tid & 31;
  v8f accR[3], accI[3];
  int mB[3], nB[3], nAss = 0;
  for (int t = wave; t < TOT; t += 8){
    mB[nAss] = (t / NT)*16; nB[nAss] = (t - (t/NT)*NT)*16;
    nAss++;
  }
  #pragma unroll
  for (int s=0;s<3;s++)
    #pragma unroll
    for (int r=0;r<8;r++){ accR[s][r]=0.f; accI[s][r]=0.f; }

  __syncthreads();   // atoms ready

  for (int k0 = 0; k0 < K; k0 += 16){
    // async DMA: edge chunk (96x16 complex, 8B per lane-element) directly into LDS
    #pragma unroll
    for (int it = 0; it < 6; ++it){                 // 6*256 == 96*16 exactly
      int idx = it*256 + tid;
      int i = idx >> 4, kt = idx & 15;
      int kg = k0 + kt; int j = kg / P, p = kg - j*P;
      const float* ep = E + (((((size_t)b*96 + i)*96 + j)*64 + c)*P + p)*2;
      unsigned dst = ldsOff_(&e2[i*17+kt]);
      uint64_t src = (uint64_t)(uintptr_t)ep;
      asm volatile("global_load_async_to_lds_b64 %0, %1, off"
                   :: "v"(dst), "v"(src) : "memory");
    }
    // overlap: build T chunk (VALU from atoms already in LDS) while DMA in flight
    for (int idx = tid; idx < 16*Wp; idx += 256){
      int kt = idx / Wp, wc = idx - kt*Wp;
      float vr = 0.f, vi = 0.f;
      if (wc < W){
        int kg = k0 + kt; int j = kg / P, p = kg - j*P;
        float cf = aggCoef[wc*8+p];
        int   q  = aggQ[wc*8+p];
        vr = cf * aR[j*17+q];
        vi = cf * aI[j*17+q];
      }
      tR[kt*65+wc] = vr; tI[kt*65+wc] = vi;
    }
    asm volatile("s_wait_asynccnt 0" ::: "memory");
    __syncthreads();
    #pragma unroll
    for (int s=0;s<3;s++){
      if (s < nAss){
        const int rowA = (mB[s] + (lane & 15))*17;
        const int colB =  nB[s] + (lane & 15);
        const int kh   = (lane >> 4) << 1;
        #pragma unroll
        for (int ks=0; ks<16; ks+=4){
          int ka = ks + kh;
          float2 ev0 = e2[rowA+ka];
          float2 ev1 = e2[rowA+ka+1];
          v2f ar2, ai2, br2, bi2;
          ar2[0]=ev0.x; ar2[1]=ev1.x;
          ai2[0]=ev0.y; ai2[1]=ev1.y;
          br2[0]=tR[ka*65+colB]; br2[1]=tR[(ka+1)*65+colB];
          bi2[0]=tI[ka*65+colB]; bi2[1]=tI[(ka+1)*65+colB];
          v2f ain2 = -ai2;
          accR[s] = __builtin_amdgcn_wmma_f32_16x16x4_f32(false, ar2,  false, br2, (short)0, accR[s], false, false);
          accR[s] = __builtin_amdgcn_wmma_f32_16x16x4_f32(false, ain2, false, bi2, (short)0, accR[s], false, false);
          accI[s] = __builtin_amdgcn_wmma_f32_16x16x4_f32(false, ar2,  false, bi2, (short)0, accI[s], false, false);
          accI[s] = __builtin_amdgcn_wmma_f32_16x16x4_f32(false, ai2,  false, br2, (short)0, accI[s], false, false);
        }
      }
    }
    __syncthreads();
  }

  #pragma unroll
  for (int s=0;s<3;s++){
    if (s < nAss){
      int wc = nB[s] + (lane & 15);
      if (wc < W){
        int slot = aggMeta[wc*4+0];
        int lo   = aggMeta[wc*4+1];
        int mi   = aggMeta[wc*4+2];
        size_t t = (size_t)slot*64 + c;
        float* cr = ws + CATR[lo];
        float* cm = ws + CATI[lo];
        size_t base = (((size_t)b*cT[lo] + t)*(2*lo+1) + mi)*96 + mB[s] + ((lane>>4)<<3);
        #pragma unroll
        for (int r=0;r<8;r++){ cr[base+r] = accR[s][r]; cm[base+r] = accI[s][r]; }
      }
    }
  }
}

// ---------------- final channel mix: out = w_l @ cat, per (b,l,m) ----------------
template<int L>
__global__ __launch_bounds__(256) void mix_kernel(
    const float* __restrict__ Wt, const float* __restrict__ ws, float* __restrict__ out)
{
  constexpr int M2 = 2*L+1;
  constexpr int T  = cT[L];
  const int b  = blockIdx.x / M2;
  const int mi = blockIdx.x - b*M2;
  const float* catR = ws + CATR[L];
  const float* catI = ws + CATI[L];

  __shared__ float aR[64*17], aI[64*17];   // weights 64 x 16
  __shared__ float bR[16*97], bI[16*97];   // cat chunk 16 x 96

  const int tid = threadIdx.x;
  const int wave = tid >> 5, lane = tid & 31;
  v8f accR[3], accI[3];
  int dB[3], iB[3];
  #pragma unroll
  for (int s=0;s<3;s++){
    int t = wave + s*8;                    // 24 tiles: 4 d-tiles x 6 i-tiles
    dB[s] = (t / 6)*16; iB[s] = (t - (t/6)*6)*16;
    #pragma unroll
    for (int r=0;r<8;r++){ accR[s][r]=0.f; accI[s][r]=0.f; }
  }

  for (int k0=0; k0<T; k0+=16){
    for (int idx=tid; idx<64*16; idx+=256){
      int d = idx >> 4, kt = idx & 15;
      const float* wp = Wt + ((size_t)d*T + k0 + kt)*2;
      aR[d*17+kt] = wp[0]; aI[d*17+kt] = wp[1];
    }
    for (int idx=tid; idx<16*96; idx+=256){
      int kt = idx / 96, i = idx - kt*96;
      size_t base = (((size_t)b*T + k0 + kt)*M2 + mi)*96 + i;
      bR[kt*97+i] = catR[base]; bI[kt*97+i] = catI[base];
    }
    __syncthreads();
    #pragma unroll
    for (int s=0;s<3;s++){
      const int rowA = (dB[s] + (lane & 15))*17;
      const int colB =  iB[s] + (lane & 15);
      const int kh   = (lane >> 4) << 1;
      #pragma unroll
      for (int ks=0; ks<16; ks+=4){
        int ka = ks + kh;
        v2f ar2, ai2, br2, bi2;
        ar2[0]=aR[rowA+ka]; ar2[1]=aR[rowA+ka+1];
        ai2[0]=aI[rowA+ka]; ai2[1]=aI[rowA+ka+1];
        br2[0]=bR[ka*97+colB]; br2[1]=bR[(ka+1)*97+colB];
        bi2[0]=bI[ka*97+colB]; bi2[1]=bI[(ka+1)*97+colB];
        v2f ain2 = -ai2;
        accR[s] = __builtin_amdgcn_wmma_f32_16x16x4_f32(false, ar2,  false, br2, (short)0, accR[s], false, false);
        accR[s] = __builtin_amdgcn_wmma_f32_16x16x4_f32(false, ain2, false, bi2, (short)0, accR[s], false, false);
        accI[s] = __builtin_amdgcn_wmma_f32_16x16x4_f32(false, ar2,  false, bi2, (short)0, accI[s], false, false);
        accI[s] = __builtin_amdgcn_wmma_f32_16x16x4_f32(false, ai2,  false, br2, (short)0, accI[s], false, false);
      }
    }
    __syncthreads();
  }

  float* op = out + OUT_OFF[L];
  #pragma unroll
  for (int s=0;s<3;s++){
    int i  = iB[s] + (lane & 15);
    int d0 = dB[s] + ((lane>>4)<<3);
    #pragma unroll
    for (int r=0;r<8;r++){
      size_t o = ((((size_t)b*96 + i)*64 + (d0+r))*M2 + mi)*2;
      op[o]   = accR[s][r];
      op[o+1] = accI[s][r];
    }
  }
}

// ---------------- launcher ----------------
extern "C" void kernel_launch(void* const* d_in, const int* in_sizes, int n_in,
                              void* d_out, int out_size, void* d_ws, size_t ws_size,
                              hipStream_t stream) {
  (void)in_sizes; (void)n_in; (void)out_size;
  if (ws_size < WS_FLOATS * sizeof(float)) return;

  // setup_inputs dict order is interleaved: atom{l}, edge{l}, w{l} per l, then mask
  const float* A[4]; const float* E[4]; const float* Wp[4];
  for (int l=0;l<4;l++){
    A[l]  = (const float*)d_in[3*l + 0];
    E[l]  = (const float*)d_in[3*l + 1];
    Wp[l] = (const float*)d_in[3*l + 2];
  }
  float* ws  = (float*)d_ws;
  float* out = (float*)d_out;

  init_plan_kernel<<<1, 64, 0, stream>>>(ws);
  sq_kernel<<<96, 256, 0, stream>>>(A[0], A[1], A[2], A[3], ws);

  agg_kernel<0><<<256, 256, 0, stream>>>(E[0], A[0], A[1], A[2], A[3], ws);
  agg_kernel<1><<<256, 256, 0, stream>>>(E[1], A[0], A[1], A[2], A[3], ws);
  agg_kernel<2><<<256, 256, 0, stream>>>(E[2], A[0], A[1], A[2], A[3], ws);
  agg_kernel<3><<<256, 256, 0, stream>>>(E[3], A[0], A[1], A[2], A[3], ws);

  mix_kernel<0><<<4*1,  256, 0, stream>>>(Wp[0], ws, out);
  mix_kernel<1><<<4*3,  256, 0, stream>>>(Wp[1], ws, out);
  mix_kernel<2><<<4*5,  256, 0, stream>>>(Wp[2], ws, out);
  mix_kernel<3><<<4*7,  256, 0, stream>>>(Wp[3], ws, out);
}